// Policy_AE_63574105915522
// MI455X (gfx1250) — compile-verified
//
#include <hip/hip_runtime.h>
#include <stdint.h>

// ---------------------------------------------------------------------------
// Policy AE on MI455X (gfx1250, wave32).
// All matrix math runs on V_WMMA_F32_16X16X4_F32 (native f32 matrix pipe).
// Fragment layouts (per CDNA5 ISA 7.12.2, 32-bit A 16x4):
//   A frag: lane l holds A[M = l%16][K = k + 2*(l/16) + {0,1}] in 2 VGPRs
//   B frag: lane l holds B[K = k + 2*(l/16) + {0,1}][N = l%16]
//   C/D:    VGPR j, lane l -> row (j + 8*(l/16)), col (l%16)
// ---------------------------------------------------------------------------

typedef __attribute__((ext_vector_type(2))) float v2f;
typedef __attribute__((ext_vector_type(8))) float v8f;

#define N_NODES 12288
#define D_INF   512
#define H_DIM   256
#define D_OUTF  64
#define E_EDGES 393216
#define EPSV    1e-8f
#define INV_EPSV 1e8f

__device__ __forceinline__ v8f wmma_f32(v2f a, v2f b, v8f c) {
  // 8 args: (neg_a, A, neg_b, B, c_mod, C, reuse_a, reuse_b)
  return __builtin_amdgcn_wmma_f32_16x16x4_f32(false, a, false, b, (short)0, c,
                                               false, false);
}

__device__ __forceinline__ float fast_sigmoid(float x) {
  // 1/(1+exp(-x)) with hardware v_exp_f32 + v_rcp_f32 (no IEEE div sequence)
  return __builtin_amdgcn_rcpf(1.0f + __expf(-x));
}

// ---------------------------------------------------------------------------
// Generic fused GEMM: OUT[M x Nout] = act(A[M x KTOT] @ W[KTOT x Nout] + bias)
// Block tile 128x64, K staged in 32-wide LDS chunks. 8 waves; wave w computes
// rows [w*16, w*16+16) x all 64 cols (4 accumulators of v8f).
// LDS pitch 36 floats -> 16B-aligned b128 stores, conflict-free b64 frag reads.
// ---------------------------------------------------------------------------
template <int KTOT, bool RELU>
__global__ __launch_bounds__(256) void gemm_bias_act(
    const float* __restrict__ A, const float* __restrict__ W,
    const float* __restrict__ bias, float* __restrict__ OUT, int Nout) {
  constexpr int BM = 128, BN = 64, BK = 32, LD = 36;
  __shared__ float As[BM * LD];
  __shared__ float Wt[BN * LD];  // W tile stored transposed: Wt[n][k]

  const int tid = threadIdx.x;
  const int lane = tid & 31, wave = tid >> 5;
  const int m0 = blockIdx.y * BM;
  const int n0 = blockIdx.x * BN;
  const int lm = lane & 15;            // matrix row/col within fragment
  const int lk = (lane >> 4) << 1;     // k sub-offset: 0 or 2

  v8f acc[4] = {};

  for (int k0 = 0; k0 < KTOT; k0 += BK) {
    // Stage A tile (128x32): 1024 float4, 4 per thread, coalesced.
#pragma unroll
    for (int it = 0; it < 4; ++it) {
      int idx = it * 256 + tid;
      int r = idx >> 3, c4 = (idx & 7) << 2;
      float4 v = *(const float4*)(A + (size_t)(m0 + r) * KTOT + k0 + c4);
      *(float4*)(&As[r * LD + c4]) = v;
    }
    // Stage W tile (32x64) transposed into Wt[64][32].
#pragma unroll
    for (int it = 0; it < 2; ++it) {
      int idx = it * 256 + tid;
      int r = idx >> 4, c4 = (idx & 15) << 2;
      float4 v = *(const float4*)(W + (size_t)(k0 + r) * Nout + n0 + c4);
      Wt[(c4 + 0) * LD + r] = v.x;
      Wt[(c4 + 1) * LD + r] = v.y;
      Wt[(c4 + 2) * LD + r] = v.z;
      Wt[(c4 + 3) * LD + r] = v.w;
    }
    __syncthreads();
#pragma unroll
    for (int kk = 0; kk < BK; kk += 4) {
      v2f a = *(const v2f*)(&As[(wave * 16 + lm) * LD + kk + lk]);
#pragma unroll
      for (int t = 0; t < 4; ++t) {
        v2f b = *(const v2f*)(&Wt[(t * 16 + lm) * LD + kk + lk]);
        acc[t] = wmma_f32(a, b, acc[t]);
      }
    }
    __syncthreads();
  }

  // Epilogue: bias + optional ReLU, coalesced 16-wide row stores.
  const int rbase = m0 + wave * 16 + ((lane >> 4) << 3);
#pragma unroll
  for (int t = 0; t < 4; ++t) {
    int col = n0 + t * 16 + lm;
    float bv = bias[col];
#pragma unroll
    for (int j = 0; j < 8; ++j) {
      float v = acc[t][j] + bv;
      if (RELU) v = fmaxf(v, 0.0f);
      OUT[(size_t)(rbase + j) * Nout + col] = v;
    }
  }
}

// ---------------------------------------------------------------------------
// Row L2 norms of z [N x 64] -> reciprocal norms.
// rin[i] = 1/||z_i||  (precise; only N divides total). Downstream uses
// 1/max(n_i*n_j, eps) == min(rin_i*rin_j, 1/eps) for non-negative norms
// (||z||=0 -> rin=inf -> clamped to 1/eps, matching the reference).
// ---------------------------------------------------------------------------
__global__ __launch_bounds__(256) void row_rnorm(const float* __restrict__ z,
                                                 float* __restrict__ rin) {
  int i = blockIdx.x * 256 + threadIdx.x;
  const float4* p = (const float4*)(z + (size_t)i * D_OUTF);
  float s = 0.0f;
#pragma unroll
  for (int q = 0; q < 16; ++q) {
    float4 v = p[q];
    s += v.x * v.x + v.y * v.y + v.z * v.z + v.w * v.w;
  }
  rin[i] = 1.0f / sqrtf(s);
}

// ---------------------------------------------------------------------------
// Gram + cosine + sigmoid: out[i][j] = sigmoid((z_i . z_j)/max(|z_i||z_j|,eps))
// Block tile 128x128, full K=64 staged once (z rows are L2-resident anyway).
// Wave w: rows [w*16, w*16+16) x 128 cols -> 8 v8f accumulators,
// 16 k-steps x 8 tiles = 128 v_wmma_f32_16x16x4_f32 per wave.
// Epilogue is div-free: mul by clamped reciprocal-norm product + exp + rcp.
// ---------------------------------------------------------------------------
__global__ __launch_bounds__(256) void gram_sigmoid(
    const float* __restrict__ z, const float* __restrict__ rin,
    float* __restrict__ out) {
  constexpr int BM = 128, LD = 68;  // 68-float pitch: 16B aligned, no conflicts
  __shared__ float Zm[BM * LD];
  __shared__ float Zn[BM * LD];

  const int tid = threadIdx.x;
  const int lane = tid & 31, wave = tid >> 5;
  const int m0 = blockIdx.y * BM;
  const int n0 = blockIdx.x * BM;
  const int lm = lane & 15;
  const int lk = (lane >> 4) << 1;

  // Stage both 128x64 z tiles: 2048 float4 each, 8 per thread.
#pragma unroll
  for (int it = 0; it < 8; ++it) {
    int idx = it * 256 + tid;
    int r = idx >> 4, c4 = (idx & 15) << 2;
    *(float4*)(&Zm[r * LD + c4]) = *(const float4*)(z + (size_t)(m0 + r) * D_OUTF + c4);
    *(float4*)(&Zn[r * LD + c4]) = *(const float4*)(z + (size_t)(n0 + r) * D_OUTF + c4);
  }
  __syncthreads();

  v8f acc[8] = {};
#pragma unroll
  for (int kk = 0; kk < D_OUTF; kk += 4) {
    v2f a = *(const v2f*)(&Zm[(wave * 16 + lm) * LD + kk + lk]);
#pragma unroll
    for (int t = 0; t < 8; ++t) {
      v2f b = *(const v2f*)(&Zn[(t * 16 + lm) * LD + kk + lk]);
      acc[t] = wmma_f32(a, b, acc[t]);
    }
  }

  const int rbase = m0 + wave * 16 + ((lane >> 4) << 3);
  float rr[8];
#pragma unroll
  for (int j = 0; j < 8; ++j) rr[j] = rin[rbase + j];

#pragma unroll
  for (int t = 0; t < 8; ++t) {
    int col = n0 + t * 16 + lm;
    float rc = rin[col];
#pragma unroll
    for (int j = 0; j < 8; ++j) {
      float scale = fminf(rr[j] * rc, INV_EPSV);
      out[(size_t)(rbase + j) * N_NODES + col] = fast_sigmoid(acc[t][j] * scale);
    }
  }
}

// ---------------------------------------------------------------------------
// Edge loss, stage 1: per-block partial sums of softplus(-cos_sim).
// z (3 MB) and rin are L2-resident, so the random gathers hit cache.
// ---------------------------------------------------------------------------
__global__ __launch_bounds__(256) void edge_loss_partial(
    const float* __restrict__ z, const float* __restrict__ rin,
    const long long* __restrict__ ei, float* __restrict__ partial) {
  __shared__ float red[256];
  int e = blockIdx.x * 256 + threadIdx.x;
  int s = (int)ei[e];
  int d = (int)ei[E_EDGES + e];
  const float4* zs = (const float4*)(z + (size_t)s * D_OUTF);
  const float4* zd = (const float4*)(z + (size_t)d * D_OUTF);
  float dot = 0.0f;
#pragma unroll
  for (int q = 0; q < 16; ++q) {
    float4 a = zs[q], b = zd[q];
    dot += a.x * b.x + a.y * b.y + a.z * b.z + a.w * b.w;
  }
  float sim = dot * fminf(rin[s] * rin[d], INV_EPSV);
  // -log(sigmoid(sim)) = softplus(-sim), numerically stable form
  float t = -sim;
  float contrib = fmaxf(t, 0.0f) + log1pf(__expf(-fabsf(t)));
  red[threadIdx.x] = contrib;
  __syncthreads();
  for (int off = 128; off > 0; off >>= 1) {
    if (threadIdx.x < off) red[threadIdx.x] += red[threadIdx.x + off];
    __syncthreads();
  }
  if (threadIdx.x == 0) partial[blockIdx.x] = red[0];
}

// Stage 2: deterministic fixed-order final reduction of 1536 partials.
__global__ __launch_bounds__(256) void edge_loss_final(
    const float* __restrict__ partial, float* __restrict__ out_loss) {
  __shared__ float red[256];
  float s = 0.0f;
  for (int i = threadIdx.x; i < (E_EDGES / 256); i += 256) s += partial[i];
  red[threadIdx.x] = s;
  __syncthreads();
  for (int off = 128; off > 0; off >>= 1) {
    if (threadIdx.x < off) red[threadIdx.x] += red[threadIdx.x + off];
    __syncthreads();
  }
  if (threadIdx.x == 0) out_loss[0] = red[0] / (float)E_EDGES;
}

// ---------------------------------------------------------------------------
extern "C" void kernel_launch(void* const* d_in, const int* in_sizes, int n_in,
                              void* d_out, int out_size, void* d_ws,
                              size_t ws_size, hipStream_t stream) {
  (void)in_sizes; (void)n_in; (void)out_size; (void)ws_size;
  const float* x  = (const float*)d_in[0];
  const long long* ei = (const long long*)d_in[1];  // reference: int64
  const float* W0 = (const float*)d_in[2];
  const float* b0 = (const float*)d_in[3];
  const float* W1 = (const float*)d_in[4];
  const float* b1 = (const float*)d_in[5];
  const float* W2 = (const float*)d_in[6];
  const float* b2 = (const float*)d_in[7];
  float* out = (float*)d_out;  // [N*N] sim_matrix, then [1] loss

  float* ws = (float*)d_ws;
  float* h0 = ws;                                   // N*H
  float* h1 = h0 + (size_t)N_NODES * H_DIM;         // N*H
  float* zb = h1 + (size_t)N_NODES * H_DIM;         // N*D_OUT
  float* ri = zb + (size_t)N_NODES * D_OUTF;        // N (reciprocal norms)
  float* pp = ri + N_NODES;                         // E/256 partials

  dim3 blk(256);
  // h0 = relu(x @ W0 + b0)
  gemm_bias_act<D_INF, true>
      <<<dim3(H_DIM / 64, N_NODES / 128), blk, 0, stream>>>(x, W0, b0, h0, H_DIM);
  // h1 = relu(h0 @ W1 + b1)
  gemm_bias_act<H_DIM, true>
      <<<dim3(H_DIM / 64, N_NODES / 128), blk, 0, stream>>>(h0, W1, b1, h1, H_DIM);
  // z = h1 @ W2 + b2
  gemm_bias_act<H_DIM, false>
      <<<dim3(D_OUTF / 64, N_NODES / 128), blk, 0, stream>>>(h1, W2, b2, zb, D_OUTF);
  // reciprocal row norms
  row_rnorm<<<N_NODES / 256, blk, 0, stream>>>(zb, ri);
  // sim_matrix = sigmoid((z z^T) * min(outer(rin,rin), 1/eps))
  gram_sigmoid<<<dim3(N_NODES / 128, N_NODES / 128), blk, 0, stream>>>(zb, ri, out);
  // loss_recon = mean(softplus(-cos_sim(edge)))
  edge_loss_partial<<<E_EDGES / 256, blk, 0, stream>>>(zb, ri, ei, pp);
  edge_loss_final<<<1, blk, 0, stream>>>(pp, out + (size_t)N_NODES * N_NODES);
}